// MatsimGNN_40742059770443
// MI455X (gfx1250) — compile-verified
//
#include <hip/hip_runtime.h>

typedef float v2f __attribute__((ext_vector_type(2)));
typedef float v8f __attribute__((ext_vector_type(8)));

// ---------------- degree / norm ----------------

__global__ void k_init_deg(int* __restrict__ deg, int n) {
    int i = blockIdx.x * blockDim.x + threadIdx.x;
    if (i < n) deg[i] = 1;                       // self-loop
}

__global__ void k_deg_accum(const int* __restrict__ dst, int* __restrict__ deg, int e) {
    int i = blockIdx.x * blockDim.x + threadIdx.x;
    if (i < e) atomicAdd(&deg[dst[i]], 1);
}

// in-place: reinterpret int degree -> float rsqrt(degree)
__global__ void k_rsqrt(int* __restrict__ degi, float* __restrict__ dis, int n) {
    int i = blockIdx.x * blockDim.x + threadIdx.x;
    if (i < n) {
        float d = (float)degi[i];                // always >= 1
        dis[i] = rsqrtf(d);
    }
}

// ---------------- dense transform via f32 WMMA (16x16x4) ----------------
// One wave computes a 16x16 tile of T = X @ W.  N must be a multiple of 16
// (N = 100000 = 6250*16).  Full-f32 precision.

template <int CIN, int COUT>
__global__ __launch_bounds__(256) void k_gemm_wmma(const float* __restrict__ X,
                                                   const float* __restrict__ W,
                                                   float* __restrict__ T, int n) {
    const int lane        = threadIdx.x & 31;
    const int waveInBlk   = threadIdx.x >> 5;
    const int wavesPerBlk = blockDim.x >> 5;
    const int colTiles    = COUT / 16;
    const int rowTiles    = n / 16;
    const int tile        = blockIdx.x * wavesPerBlk + waveInBlk;
    if (tile >= rowTiles * colTiles) return;     // wave-uniform: EXEC stays all-1s

    const int rt    = tile / colTiles;
    const int ct    = tile % colTiles;
    const int row0  = rt * 16;
    const int col0  = ct * 16;
    const int mrow  = lane & 15;                 // row (A) / col (B,C,D) within tile
    const int khalf = (lane >> 4) << 1;          // lanes 16-31 hold K+2,K+3

    v8f c = {};
#pragma unroll
    for (int k0 = 0; k0 < CIN; k0 += 4) {
        // A fragment: 16x4 f32; lane holds {K=k0+khalf, K=k0+khalf+1} of row mrow
        v2f a = *(const v2f*)(X + (size_t)(row0 + mrow) * CIN + k0 + khalf);
        // B fragment: 4x16 f32; lane holds rows {k0+khalf, k0+khalf+1} of col mrow
        v2f b;
        b.x = W[(k0 + khalf + 0) * COUT + col0 + mrow];
        b.y = W[(k0 + khalf + 1) * COUT + col0 + mrow];
        c = __builtin_amdgcn_wmma_f32_16x16x4_f32(false, a, false, b,
                                                  (short)0, c, false, false);
    }
    // C/D layout: VGPR i -> M = i + (lane>=16 ? 8 : 0), N = lane&15
#pragma unroll
    for (int i = 0; i < 8; ++i) {
        int r = row0 + i + ((lane >> 4) << 3);
        T[(size_t)r * COUT + col0 + mrow] = c[i];
    }
}

// ---------------- edge scatter-add (message passing) ----------------

template <int C>
__global__ void k_agg(const int* __restrict__ src, const int* __restrict__ dst,
                      const float* __restrict__ dis, const float* __restrict__ T,
                      float* __restrict__ out, int e) {
    int i = blockIdx.x * blockDim.x + threadIdx.x;
    if (i >= e) return;
    int s = src[i], d = dst[i];
    float w = dis[s] * dis[d];
    const float4* ts = (const float4*)(T + (size_t)s * C);
    float* od = out + (size_t)d * C;
#pragma unroll
    for (int q = 0; q < C / 4; ++q) {
        float4 v = ts[q];
        unsafeAtomicAdd(od + 4 * q + 0, v.x * w);
        unsafeAtomicAdd(od + 4 * q + 1, v.y * w);
        unsafeAtomicAdd(od + 4 * q + 2, v.z * w);
        unsafeAtomicAdd(od + 4 * q + 3, v.w * w);
    }
}

// ---------------- epilogue: self-loop + bias + relu ----------------

template <int C>
__global__ void k_epilogue(float* __restrict__ out, const float* __restrict__ T,
                           const float* __restrict__ dis, const float* __restrict__ b,
                           int n) {
    int idx = blockIdx.x * blockDim.x + threadIdx.x;   // over n * C/4 quads
    int total = n * (C / 4);
    if (idx >= total) return;
    int i = idx / (C / 4);
    int q = idx % (C / 4);
    float d2 = dis[i];
    d2 *= d2;                                          // self-loop norm = dis^2
    float4 t  = ((const float4*)(T + (size_t)i * C))[q];
    float4 o  = ((const float4*)(out + (size_t)i * C))[q];
    float4 bb = ((const float4*)b)[q];
    o.x = fmaxf(o.x + t.x * d2 + bb.x, 0.0f);
    o.y = fmaxf(o.y + t.y * d2 + bb.y, 0.0f);
    o.z = fmaxf(o.z + t.z * d2 + bb.z, 0.0f);
    o.w = fmaxf(o.w + t.w * d2 + bb.w, 0.0f);
    ((float4*)(out + (size_t)i * C))[q] = o;
}

// ---------------- final projection + mean reduction ----------------

__global__ void k_final(const float* __restrict__ H, const float* __restrict__ Wfc,
                        const float* __restrict__ bfc, float* __restrict__ outp, int n) {
    __shared__ float sdata[256];
    int i = blockIdx.x * blockDim.x + threadIdx.x;
    float v = 0.0f;
    if (i < n) {
        float acc = 0.0f;
#pragma unroll
        for (int c = 0; c < 16; ++c) acc += H[(size_t)i * 16 + c] * Wfc[c];
        v = (acc + bfc[0]) / (float)n;
    }
    sdata[threadIdx.x] = v;
    __syncthreads();
    for (int s = blockDim.x / 2; s > 0; s >>= 1) {
        if (threadIdx.x < s) sdata[threadIdx.x] += sdata[threadIdx.x + s];
        __syncthreads();
    }
    if (threadIdx.x == 0) unsafeAtomicAdd(outp, sdata[0]);
}

// ---------------- host launch ----------------

extern "C" void kernel_launch(void* const* d_in, const int* in_sizes, int n_in,
                              void* d_out, int out_size, void* d_ws, size_t ws_size,
                              hipStream_t stream) {
    const float* x   = (const float*)d_in[0];
    const int*   ei  = (const int*)d_in[1];
    const float* W1  = (const float*)d_in[2];
    const float* b1  = (const float*)d_in[3];
    const float* W2  = (const float*)d_in[4];
    const float* b2  = (const float*)d_in[5];
    const float* W3  = (const float*)d_in[6];
    const float* b3  = (const float*)d_in[7];
    const float* Wfc = (const float*)d_in[8];
    const float* bfc = (const float*)d_in[9];
    float* out = (float*)d_out;

    const int N = in_sizes[0] / 32;   // 100000
    const int E = in_sizes[1] / 2;    // 3200000
    const int* src = ei;              // edge_index[0]
    const int* dst = ei + E;          // edge_index[1]

    float* wsf  = (float*)d_ws;
    float* dis  = wsf;                              // N floats (also int degree)
    float* bufA = wsf + N;                          // N*32 floats
    float* bufB = bufA + (size_t)N * 32;            // N*32 floats
    float* bufC = bufB + (size_t)N * 32;            // N*32 floats
    int*   degi = (int*)dis;

    const int TB = 256;
    dim3 blkN((N + TB - 1) / TB), blkE((E + TB - 1) / TB), thr(TB);

    // symmetric normalization: deg (int, deterministic) -> dis = rsqrt(deg)
    k_init_deg<<<blkN, thr, 0, stream>>>(degi, N);
    k_deg_accum<<<blkE, thr, 0, stream>>>(dst, degi, E);
    k_rsqrt<<<blkN, thr, 0, stream>>>(degi, dis, N);

    // ---- layer 1: x(32) -> t=bufA(16) -> h1=bufB(16)
    {
        int tiles = (N / 16) * (16 / 16);
        k_gemm_wmma<32, 16><<<(tiles + 7) / 8, thr, 0, stream>>>(x, W1, bufA, N);
        hipMemsetAsync(bufB, 0, (size_t)N * 16 * sizeof(float), stream);
        k_agg<16><<<blkE, thr, 0, stream>>>(src, dst, dis, bufA, bufB, E);
        int tot = N * 4;
        k_epilogue<16><<<(tot + TB - 1) / TB, thr, 0, stream>>>(bufB, bufA, dis, b1, N);
    }
    // ---- layer 2: h1=bufB(16) -> t=bufC(32) -> h2=bufA(32)
    {
        int tiles = (N / 16) * (32 / 16);
        k_gemm_wmma<16, 32><<<(tiles + 7) / 8, thr, 0, stream>>>(bufB, W2, bufC, N);
        hipMemsetAsync(bufA, 0, (size_t)N * 32 * sizeof(float), stream);
        k_agg<32><<<blkE, thr, 0, stream>>>(src, dst, dis, bufC, bufA, E);
        int tot = N * 8;
        k_epilogue<32><<<(tot + TB - 1) / TB, thr, 0, stream>>>(bufA, bufC, dis, b2, N);
    }
    // ---- layer 3: h2=bufA(32) -> t=bufB(16) -> h3=bufC(16)
    {
        int tiles = (N / 16) * (16 / 16);
        k_gemm_wmma<32, 16><<<(tiles + 7) / 8, thr, 0, stream>>>(bufA, W3, bufB, N);
        hipMemsetAsync(bufC, 0, (size_t)N * 16 * sizeof(float), stream);
        k_agg<16><<<blkE, thr, 0, stream>>>(src, dst, dis, bufB, bufC, E);
        int tot = N * 4;
        k_epilogue<16><<<(tot + TB - 1) / TB, thr, 0, stream>>>(bufC, bufB, dis, b3, N);
    }

    // ---- mean(h3 @ Wfc + bfc)
    hipMemsetAsync(out, 0, sizeof(float), stream);
    k_final<<<blkN, thr, 0, stream>>>(bufC, Wfc, bfc, out, N);
}